// Voxelizer_75642964017640
// MI455X (gfx1250) — compile-verified
//
#include <hip/hip_runtime.h>
#include <stdint.h>

// ---------------------------------------------------------------------------
// Voxelizer for MI455X (gfx1250).
// Memory-bound scatter-reduce: ~6 MB traffic, ~5 MFLOP -> HBM/L2-atomic bound.
// CDNA5 paths used: global_load_async_to_lds_b32 + s_wait_asynccnt +
// ds_load readback (single fused asm block so the optimizer cannot elide the
// LDS traffic), and relaxed agent-scope f32/u32 hardware atomics (resolved in
// the 192 MB L2, where the whole 1.1 MB accumulator lives).
// ---------------------------------------------------------------------------

#define VGRID   24
#define VGRID3  (VGRID * VGRID * VGRID)   // 13824 voxels per batch
#define VOXSZ   0.05f
#define NACC    10                        // cnt, sx,sy,sz, sxx,sxy,sxz,syy,syz,szz
#define NBATCH  2

// Order-preserving map: float -> uint so that unsigned-min == float-min.
__device__ __forceinline__ unsigned f2key(float f) {
    unsigned u = __float_as_uint(f);
    return u ^ ((unsigned)((int)u >> 31) | 0x80000000u);
}
__device__ __forceinline__ float key2f(unsigned k) {
    unsigned u = (k & 0x80000000u) ? (k ^ 0x80000000u) : ~k;
    return __uint_as_float(u);
}

__device__ __forceinline__ int vclamp(int v) {
    return v < 0 ? 0 : (v > VGRID - 1 ? VGRID - 1 : v);
}

__device__ __forceinline__ void atomic_add_f32(float* p, float v) {
    (void)__hip_atomic_fetch_add(p, v, __ATOMIC_RELAXED, __HIP_MEMORY_SCOPE_AGENT);
}

// ---------------------------------------------------------------------------
// K0: initialize workspace (min keys = +inf key, accumulators = 0).
// Word layout of d_ws: [0..5] = min keys (B*3), [6..15] pad, [16..) = acc.
// ---------------------------------------------------------------------------
__global__ void init_ws(unsigned* __restrict__ w, int nwords) {
    int i = blockIdx.x * blockDim.x + threadIdx.x;
    int stride = gridDim.x * blockDim.x;
    for (; i < nwords; i += stride)
        w[i] = (i < NBATCH * 3) ? 0xFFFFFFFFu : 0u;
}

// ---------------------------------------------------------------------------
// K1: exact per-batch component-wise min via key mapping + atomic_min_u32.
// gridDim.y = batch. Deterministic (min is order-independent).
// ---------------------------------------------------------------------------
__global__ void batch_min(const float* __restrict__ x, unsigned* __restrict__ minkeys, int N) {
    const int b = blockIdx.y;
    const float* xb = x + (size_t)b * N * 3;
    unsigned k0 = 0xFFFFFFFFu, k1 = 0xFFFFFFFFu, k2 = 0xFFFFFFFFu;
    int stride = gridDim.x * blockDim.x;
    for (int i = blockIdx.x * blockDim.x + threadIdx.x; i < N; i += stride) {
        unsigned a = f2key(xb[(size_t)i * 3 + 0]);
        unsigned c = f2key(xb[(size_t)i * 3 + 1]);
        unsigned d = f2key(xb[(size_t)i * 3 + 2]);
        k0 = a < k0 ? a : k0;
        k1 = c < k1 ? c : k1;
        k2 = d < k2 ? d : k2;
    }
    (void)__hip_atomic_fetch_min(&minkeys[b * 3 + 0], k0, __ATOMIC_RELAXED, __HIP_MEMORY_SCOPE_AGENT);
    (void)__hip_atomic_fetch_min(&minkeys[b * 3 + 1], k1, __ATOMIC_RELAXED, __HIP_MEMORY_SCOPE_AGENT);
    (void)__hip_atomic_fetch_min(&minkeys[b * 3 + 2], k2, __ATOMIC_RELAXED, __HIP_MEMORY_SCOPE_AGENT);
}

// ---------------------------------------------------------------------------
// K2: per-voxel moment accumulation. One point per thread.
// Full blocks stage their points via CDNA5 async global->LDS copies:
// each wave moves 32 points (384 B) with 3 coalesced 128 B async transfers,
// waits on ASYNCcnt, then reads AoS back with ds_load_b32 — all inside ONE
// asm block so the LDS data flow cannot be optimized away.
// ---------------------------------------------------------------------------
__global__ void accumulate(const float* __restrict__ x,
                           const unsigned* __restrict__ minkeys,
                           float* __restrict__ acc,
                           int N, int total) {
    __shared__ float lds[768];               // 256 threads * 3 floats
    const int t = blockIdx.x * 256 + threadIdx.x;
    float px, py, pz;

    if ((blockIdx.x + 1) * 256 <= total) {
        const int wave = threadIdx.x >> 5;
        const int lane = threadIdx.x & 31;
        const int waveStart = blockIdx.x * 256 + wave * 32;     // first point of wave
        const float* src = x + (size_t)waveStart * 3;           // 96 consecutive floats
        // Byte offsets of this lane's LDS slots (low 32 bits of the generic
        // shared-aperture address == offset from the workgroup LDS base).
        const unsigned l0    = (unsigned)(uintptr_t)&lds[wave * 96 + lane];
        const unsigned lread = (unsigned)(uintptr_t)&lds[wave * 96 + lane * 3];
        asm volatile(
            "global_load_async_to_lds_b32 %3, %4, off\n\t"
            "global_load_async_to_lds_b32 %5, %6, off\n\t"
            "global_load_async_to_lds_b32 %7, %8, off\n\t"
            "s_wait_asynccnt 0\n\t"
            "ds_load_b32 %0, %9\n\t"
            "ds_load_b32 %1, %9 offset:4\n\t"
            "ds_load_b32 %2, %9 offset:8\n\t"
            "s_wait_dscnt 0"
            : "=&v"(px), "=&v"(py), "=&v"(pz)
            : "v"(l0),          "v"(src + lane),
              "v"(l0 + 128u),   "v"(src + 32 + lane),
              "v"(l0 + 256u),   "v"(src + 64 + lane),
              "v"(lread)
            : "memory");
    } else {
        if (t >= total) return;
        const float* p = x + (size_t)t * 3;
        px = p[0]; py = p[1]; pz = p[2];
    }

    const int b = t / N;
    const float mx = key2f(minkeys[b * 3 + 0]);
    const float my = key2f(minkeys[b * 3 + 1]);
    const float mz = key2f(minkeys[b * 3 + 2]);
    const int ix = vclamp((int)floorf((px - mx) / VOXSZ));
    const int iy = vclamp((int)floorf((py - my) / VOXSZ));
    const int iz = vclamp((int)floorf((pz - mz) / VOXSZ));
    const int lin = (ix * VGRID + iy) * VGRID + iz;
    float* a = acc + (size_t)(b * VGRID3 + lin) * NACC;

    atomic_add_f32(a + 0, 1.0f);
    atomic_add_f32(a + 1, px);
    atomic_add_f32(a + 2, py);
    atomic_add_f32(a + 3, pz);
    atomic_add_f32(a + 4, px * px);
    atomic_add_f32(a + 5, px * py);
    atomic_add_f32(a + 6, px * pz);
    atomic_add_f32(a + 7, py * py);
    atomic_add_f32(a + 8, py * pz);
    atomic_add_f32(a + 9, pz * pz);
}

// ---------------------------------------------------------------------------
// K3: gather sampled points, recompute identical voxel index, finalize
// mean + covariance, emit (B, D, 12) = [mean(3), cov(9, row-major, symmetric)].
// ---------------------------------------------------------------------------
__global__ void gather_finalize(const float* __restrict__ x,
                                const int* __restrict__ sidx,
                                const unsigned* __restrict__ minkeys,
                                const float* __restrict__ acc,
                                float* __restrict__ out,
                                int N, int D, int BD) {
    const int t = blockIdx.x * blockDim.x + threadIdx.x;
    if (t >= BD) return;
    const int b = t / D;
    const int i = sidx[t];
    const float* p = x + ((size_t)b * N + i) * 3;
    const float px = p[0], py = p[1], pz = p[2];

    const float mcx = key2f(minkeys[b * 3 + 0]);
    const float mcy = key2f(minkeys[b * 3 + 1]);
    const float mcz = key2f(minkeys[b * 3 + 2]);
    const int ix = vclamp((int)floorf((px - mcx) / VOXSZ));
    const int iy = vclamp((int)floorf((py - mcy) / VOXSZ));
    const int iz = vclamp((int)floorf((pz - mcz) / VOXSZ));
    const int lin = (ix * VGRID + iy) * VGRID + iz;
    const float* a = acc + (size_t)(b * VGRID3 + lin) * NACC;

    const float c  = fmaxf(a[0], 1.0f);
    const float mx = a[1] / c, my = a[2] / c, mz = a[3] / c;
    const float cxx = a[4] / c - mx * mx;
    const float cxy = a[5] / c - mx * my;
    const float cxz = a[6] / c - mx * mz;
    const float cyy = a[7] / c - my * my;
    const float cyz = a[8] / c - my * mz;
    const float czz = a[9] / c - mz * mz;

    float* o = out + (size_t)t * 12;
    o[0] = mx;  o[1] = my;  o[2] = mz;
    o[3] = cxx; o[4] = cxy; o[5] = cxz;
    o[6] = cxy; o[7] = cyy; o[8] = cyz;
    o[9] = cxz; o[10] = cyz; o[11] = czz;
}

// ---------------------------------------------------------------------------
extern "C" void kernel_launch(void* const* d_in, const int* in_sizes, int n_in,
                              void* d_out, int out_size, void* d_ws, size_t ws_size,
                              hipStream_t stream) {
    const float* x    = (const float*)d_in[0];
    const int*   sidx = (const int*)d_in[1];
    float*       out  = (float*)d_out;

    const int B = NBATCH;
    const int N = in_sizes[0] / (B * 3);   // points per batch
    const int D = in_sizes[1] / B;         // samples per batch
    const int total = B * N;

    unsigned* minkeys = (unsigned*)d_ws;
    float*    acc     = (float*)d_ws + 16;             // 64 B aligned past keys
    const int accN    = B * VGRID3 * NACC;             // 276480 floats (~1.1 MB)
    const int nwords  = 16 + accN;

    // K0: init workspace
    init_ws<<<(nwords + 255) / 256, 256, 0, stream>>>(minkeys, nwords);

    // K1: per-batch mins (32 blocks x 256 threads per batch, ~16 pts/thread)
    dim3 gmin(32, B);
    batch_min<<<gmin, 256, 0, stream>>>(x, minkeys, N);

    // K2: scatter moments (one point per thread, async-staged through LDS)
    accumulate<<<(total + 255) / 256, 256, 0, stream>>>(x, minkeys, acc, N, total);

    // K3: gather + finalize sampled voxels
    const int BD = B * D;
    gather_finalize<<<(BD + 255) / 256, 256, 0, stream>>>(x, sidx, minkeys, acc, out, N, D, BD);
}